// HybridMultiHeadLinearAttention_36996848288348
// MI455X (gfx1250) — compile-verified
//
#include <hip/hip_runtime.h>
#include <hip/hip_bf16.h>

#define S_LEN 2048
#define B_SZ  16
#define E_DIM 1024
#define HD_DIM 64
#define NL_H  8
#define NG_H  8
#define T_TOK (S_LEN * B_SZ)
#define ROWSTRIDE (B_SZ * E_DIM)   // stride in floats between consecutive s for fixed (b,e)

typedef float v2f __attribute__((ext_vector_type(2)));
typedef float v8f __attribute__((ext_vector_type(8)));
typedef int   v4i __attribute__((ext_vector_type(4)));

// ------------------------------------------------------------------
// Async global->LDS copy helpers (CDNA5 GLOBAL_LOAD_ASYNC_TO_LDS_B128,
// tracked with ASYNCcnt). The builtin takes AS(1)/AS(3) int4* pointers
// (per hipcc diagnostic). Guarded so the file still compiles with a
// plain b128 copy if the toolchain lacks the builtins.
// ------------------------------------------------------------------
#if __has_builtin(__builtin_amdgcn_global_load_async_to_lds_b128)
#define HAVE_ASYNC_LDS 1
#else
#define HAVE_ASYNC_LDS 0
#endif

__device__ __forceinline__ void lds_copy_b128(const float* g, float* l) {
#if HAVE_ASYNC_LDS
  __builtin_amdgcn_global_load_async_to_lds_b128(
      (__attribute__((address_space(1))) v4i*)(g),
      (__attribute__((address_space(3))) v4i*)(l),
      0, 0);
#else
  *reinterpret_cast<float4*>(l) = *reinterpret_cast<const float4*>(g);
#endif
}

__device__ __forceinline__ void lds_copy_wait() {
#if HAVE_ASYNC_LDS
# if __has_builtin(__builtin_amdgcn_s_wait_asynccnt)
  __builtin_amdgcn_s_wait_asynccnt(0);
# else
  asm volatile("s_wait_asynccnt 0x0" ::: "memory");
# endif
#endif
}

// ------------------------------------------------------------------
// Pass 1: local scores  sc[t,h] = x[t,:] . Wls[h,:] + bls[h]
// one wave (32 lanes) per token; Wls staged in LDS (32 KB) via async DMA
// ------------------------------------------------------------------
__global__ __launch_bounds__(256) void k_scores(const float* __restrict__ x,
                                                const float* __restrict__ Wls,
                                                const float* __restrict__ bls,
                                                float* __restrict__ sc) {
  __shared__ __align__(16) float wls_s[NL_H * E_DIM];
  const int tid = threadIdx.x;
  for (int i = tid * 4; i < NL_H * E_DIM; i += 256 * 4)
    lds_copy_b128(Wls + i, wls_s + i);
  lds_copy_wait();
  __syncthreads();

  const int wave = tid >> 5;
  const int lane = tid & 31;
  const int token = blockIdx.x * 8 + wave;
  const float* xp = x + (size_t)token * E_DIM;

  float acc[8] = {0.f, 0.f, 0.f, 0.f, 0.f, 0.f, 0.f, 0.f};
#pragma unroll
  for (int i = 0; i < 8; ++i) {
    const int eo = (i * 32 + lane) * 4;
    const float4 xv = *reinterpret_cast<const float4*>(xp + eo);
#pragma unroll
    for (int h = 0; h < 8; ++h) {
      const float4 wv = *reinterpret_cast<const float4*>(wls_s + h * E_DIM + eo);
      acc[h] += xv.x * wv.x + xv.y * wv.y + xv.z * wv.z + xv.w * wv.w;
    }
  }
#pragma unroll
  for (int off = 16; off >= 1; off >>= 1) {
#pragma unroll
    for (int h = 0; h < 8; ++h) acc[h] += __shfl_xor(acc[h], off, 32);
  }
  if (lane == 0) {
#pragma unroll
    for (int h = 0; h < 8; ++h) sc[(size_t)token * 8 + h] = acc[h] + bls[h];
  }
}

// ------------------------------------------------------------------
// Pass 2: xsum[b,e] = sum_s x[s,b,e]   (coalesced over e, loop over s)
// ------------------------------------------------------------------
__global__ __launch_bounds__(256) void k_xsum(const float* __restrict__ x,
                                              float* __restrict__ xsum) {
  const int gid = blockIdx.x * 256 + threadIdx.x;   // b*1024 + e, 0..16383
  const float* p = x + gid;
  float s = 0.f;
#pragma unroll 4
  for (int si = 0; si < S_LEN; ++si) s += p[(size_t)si * ROWSTRIDE];
  xsum[gid] = s;
}

// ------------------------------------------------------------------
// Softmax over the sequence axis for each (b,h); in-place on sc -> w
// ------------------------------------------------------------------
__global__ __launch_bounds__(256) void k_softmax_l(float* __restrict__ w) {
  __shared__ float red[256];
  const int bh = blockIdx.x;        // b*8 + h, 0..127 ; element idx(s)=s*128+bh
  const int tid = threadIdx.x;

  float m = -1e30f;
  for (int s0 = tid; s0 < S_LEN; s0 += 256) m = fmaxf(m, w[(size_t)s0 * 128 + bh]);
  red[tid] = m; __syncthreads();
  for (int st = 128; st > 0; st >>= 1) {
    if (tid < st) red[tid] = fmaxf(red[tid], red[tid + st]);
    __syncthreads();
  }
  m = red[0]; __syncthreads();

  float sum = 0.f;
  for (int s0 = tid; s0 < S_LEN; s0 += 256) {
    const float v = __expf(w[(size_t)s0 * 128 + bh] - m);
    w[(size_t)s0 * 128 + bh] = v;
    sum += v;
  }
  red[tid] = sum; __syncthreads();
  for (int st = 128; st > 0; st >>= 1) {
    if (tid < st) red[tid] += red[tid + st];
    __syncthreads();
  }
  const float inv = 1.0f / red[0];
  for (int s0 = tid; s0 < S_LEN; s0 += 256) w[(size_t)s0 * 128 + bh] *= inv;
}

// ------------------------------------------------------------------
// Global scores from mean-pooled g, softmax over the BATCH axis
// ------------------------------------------------------------------
__global__ __launch_bounds__(128) void k_gscore(const float* __restrict__ xsum,
                                                const float* __restrict__ Wgs,
                                                const float* __restrict__ bgs,
                                                float* __restrict__ wg) {
  __shared__ float scs[B_SZ * NG_H];
  const int tid = threadIdx.x;
  const int b = tid >> 3;
  const int h = tid & 7;
  const float* gp = xsum + b * E_DIM;
  const float* wp = Wgs + h * E_DIM;
  float s = 0.f;
#pragma unroll 4
  for (int e = 0; e < E_DIM; ++e) s += gp[e] * wp[e];
  const float sc = s * (1.0f / (float)S_LEN) + bgs[h];
  scs[tid] = sc;
  __syncthreads();
  float m = -1e30f;
  for (int bb = 0; bb < B_SZ; ++bb) m = fmaxf(m, scs[bb * 8 + h]);
  float sum = 0.f;
  for (int bb = 0; bb < B_SZ; ++bb) sum += __expf(scs[bb * 8 + h] - m);
  wg[tid] = __expf(sc - m) / sum;
}

__global__ void k_zero(float* __restrict__ p) {
  p[blockIdx.x * 256 + threadIdx.x] = 0.f;
}

// ------------------------------------------------------------------
// Pass 3: softmax-weighted pooling  xw[b,h,e] += sum_s w[s,b,h]*x[s,b,e]
// block = (e-chunk, s-chunk, b); weights staged in LDS via async DMA;
// results merged with GLOBAL_ATOMIC_ADD_F32.
// ------------------------------------------------------------------
__global__ __launch_bounds__(256) void k_wsum(const float* __restrict__ x,
                                              const float* __restrict__ w,
                                              float* __restrict__ xw) {
  __shared__ __align__(16) float wlds[256 * 8];
  const int tid = threadIdx.x;
  const int ec = blockIdx.x;   // 0..3
  const int sc8 = blockIdx.y;  // 0..7
  const int b = blockIdx.z;    // 0..15
  const int s0 = sc8 * 256;

  {
    const float* wp = w + (size_t)(s0 + tid) * 128 + b * 8;
    lds_copy_b128(wp,     &wlds[tid * 8]);
    lds_copy_b128(wp + 4, &wlds[tid * 8 + 4]);
    lds_copy_wait();
  }
  __syncthreads();

  const int e = ec * 256 + tid;
  const float* xp = x + (size_t)s0 * ROWSTRIDE + b * E_DIM + e;
  float a0 = 0, a1 = 0, a2 = 0, a3 = 0, a4 = 0, a5 = 0, a6 = 0, a7 = 0;
#pragma unroll 4
  for (int i = 0; i < 256; ++i) {
    __builtin_prefetch(xp + (size_t)(i + 8) * ROWSTRIDE, 0, 0);
    const float xv = xp[(size_t)i * ROWSTRIDE];
    const float4* wr = reinterpret_cast<const float4*>(&wlds[i * 8]);
    const float4 w0 = wr[0], w1 = wr[1];
    a0 += w0.x * xv; a1 += w0.y * xv; a2 += w0.z * xv; a3 += w0.w * xv;
    a4 += w1.x * xv; a5 += w1.y * xv; a6 += w1.z * xv; a7 += w1.w * xv;
  }
  float* op = xw + b * (NL_H * E_DIM) + e;
  atomicAdd(op + 0 * E_DIM, a0); atomicAdd(op + 1 * E_DIM, a1);
  atomicAdd(op + 2 * E_DIM, a2); atomicAdd(op + 3 * E_DIM, a3);
  atomicAdd(op + 4 * E_DIM, a4); atomicAdd(op + 5 * E_DIM, a5);
  atomicAdd(op + 6 * E_DIM, a6); atomicAdd(op + 7 * E_DIM, a7);
}

// ------------------------------------------------------------------
// Head projections on the matrix cores: V_WMMA_F32_16X16X4_F32
// M=16 (batch), N=64 (HD, 4 tiles of 16 -> one per wave), K=1024
// heads 0..7 local (A = pooled xw), 8..15 global (A = xsum)
// Epilogue fuses biases, S*bgp, w_g scaling, and the concat layout.
// ------------------------------------------------------------------
__global__ __launch_bounds__(128) void k_headproj(const float* __restrict__ xw,
                                                  const float* __restrict__ xsum,
                                                  const float* __restrict__ Wlp,
                                                  const float* __restrict__ blp,
                                                  const float* __restrict__ Wgp,
                                                  const float* __restrict__ bgp,
                                                  const float* __restrict__ wg,
                                                  float* __restrict__ hbuf) {
  const int head = blockIdx.x;       // 0..15 (uniform per block -> EXEC stays full)
  const int tid = threadIdx.x;
  const int nt = tid >> 5;           // d-tile 0..3
  const int lane = tid & 31;
  const int r16 = lane & 15;
  const int hi = lane >> 4;
  const bool isLocal = head < 8;

  const float* A = isLocal ? (xw + r16 * (NL_H * E_DIM) + head * E_DIM)
                           : (xsum + r16 * E_DIM);
  const float* W = isLocal ? (Wlp + (size_t)head * (E_DIM * HD_DIM))
                           : (Wgp + (size_t)(head - 8) * (E_DIM * HD_DIM));
  const int col = nt * 16 + r16;

  v8f c = {};
#pragma unroll 2
  for (int kb = 0; kb < E_DIM; kb += 4) {
    const int k0 = kb + 2 * hi;                       // A layout: lanes0-15 K={0,1}, lanes16-31 K={2,3}
    const v2f a = *reinterpret_cast<const v2f*>(A + k0);
    v2f bv;
    bv.x = W[k0 * HD_DIM + col];                      // B layout mirrors A's K split
    bv.y = W[(k0 + 1) * HD_DIM + col];
    c = __builtin_amdgcn_wmma_f32_16x16x4_f32(false, a, false, bv, (short)0, c, false, false);
  }

#pragma unroll
  for (int r = 0; r < 8; ++r) {                       // C/D: VGPR r -> M=r (lo half) / r+8 (hi half)
    const int m = r + 8 * hi;
    const int d = nt * 16 + r16;
    if (isLocal) {
      hbuf[m * 1024 + head * HD_DIM + d] = c[r] + blp[head * HD_DIM + d];
    } else {
      const int hg = head - 8;
      const float v = (c[r] + (float)S_LEN * bgp[hg * HD_DIM + d]) * wg[m * 8 + hg];
      hbuf[m * 1024 + 512 + hg * HD_DIM + d] = v;
    }
  }
}

// ------------------------------------------------------------------
// Final projection: out[16,1024] = hbuf[16,1024] @ Wo^T + bo  (WMMA)
// ------------------------------------------------------------------
__global__ __launch_bounds__(128) void k_outproj(const float* __restrict__ hbuf,
                                                 const float* __restrict__ Wo,
                                                 const float* __restrict__ bo,
                                                 float* __restrict__ out) {
  const int tid = threadIdx.x;
  const int nt = blockIdx.x * 4 + (tid >> 5);  // 0..63 output-column tiles
  const int lane = tid & 31;
  const int r16 = lane & 15;
  const int hi = lane >> 4;
  const int col = nt * 16 + r16;               // output e; B[k][n] = Wo[col*1024 + k]
  const float* A = hbuf + r16 * 1024;
  const float* Brow = Wo + (size_t)col * 1024;

  v8f c = {};
#pragma unroll 2
  for (int kb = 0; kb < 1024; kb += 4) {
    const int k0 = kb + 2 * hi;
    const v2f a = *reinterpret_cast<const v2f*>(A + k0);
    const v2f bv = *reinterpret_cast<const v2f*>(Brow + k0);
    c = __builtin_amdgcn_wmma_f32_16x16x4_f32(false, a, false, bv, (short)0, c, false, false);
  }
#pragma unroll
  for (int r = 0; r < 8; ++r) {
    const int m = r + 8 * hi;
    out[m * 1024 + col] = c[r] + bo[col];
  }
}

extern "C" void kernel_launch(void* const* d_in, const int* in_sizes, int n_in,
                              void* d_out, int out_size, void* d_ws, size_t ws_size,
                              hipStream_t stream) {
  (void)in_sizes; (void)n_in; (void)out_size; (void)ws_size;
  const float* x   = (const float*)d_in[0];
  const float* Wlp = (const float*)d_in[1];
  const float* blp = (const float*)d_in[2];
  const float* Wls = (const float*)d_in[3];
  const float* bls = (const float*)d_in[4];
  const float* Wgp = (const float*)d_in[5];
  const float* bgp = (const float*)d_in[6];
  const float* Wgs = (const float*)d_in[7];
  const float* bgs = (const float*)d_in[8];
  const float* Wo  = (const float*)d_in[9];
  const float* bo  = (const float*)d_in[10];
  float* out = (float*)d_out;

  float* ws   = (float*)d_ws;
  float* sc   = ws;                 // [T,8]  scores -> softmax weights (in place), 262144 f
  float* xsum = sc + T_TOK * 8;     // [16,1024], 16384 f
  float* xw   = xsum + B_SZ * E_DIM;       // [16,8,1024], 131072 f (zero-inited)
  float* wg   = xw + B_SZ * NL_H * E_DIM;  // [16,8], 128 f
  float* hbuf = wg + B_SZ * NG_H;          // [16,1024], 16384 f

  k_scores   <<<T_TOK / 8, 256, 0, stream>>>(x, Wls, bls, sc);
  k_xsum     <<<64, 256, 0, stream>>>(x, xsum);
  k_softmax_l<<<128, 256, 0, stream>>>(sc);
  k_gscore   <<<1, 128, 0, stream>>>(xsum, Wgs, bgs, wg);
  k_zero     <<<(B_SZ * NL_H * E_DIM) / 256, 256, 0, stream>>>(xw);
  k_wsum     <<<dim3(4, 8, 16), 256, 0, stream>>>(x, sc, xw);
  k_headproj <<<16, 128, 0, stream>>>(xw, xsum, Wlp, blp, Wgp, bgp, wg, hbuf);
  k_outproj  <<<16, 128, 0, stream>>>(hbuf, Wo, bo, out);
}